// PFGSTLoss_41472204210811
// MI455X (gfx1250) — compile-verified
//
#include <hip/hip_runtime.h>
#include <hip/hip_bf16.h>
#include <math.h>

// ---------------------------------------------------------------------------
// Problem constants (from reference)
// ---------------------------------------------------------------------------
namespace {
constexpr int B  = 2,  C  = 19, H  = 192, W = 192;
constexpr int CH = 32, KIN = 256, HC = 48, WC = 48;   // projection
constexpr int KK = 5,  DILN = 2, K2 = 25, TOPK = 8;
constexpr float INV_SIG2 = 1.0f / (30.0f * 30.0f);

constexpr int NPIX   = B * H * W;             // 73728
constexpr int NC     = HC * WC;               // 2304 coarse pixels / batch
// workspace layout (floats)
constexpr int PROB_OFF = 0;
constexpr int PROB_SZ  = B * C * H * W;       // 1,400,832
constexpr int SRCP_OFF = PROB_OFF + PROB_SZ;
constexpr int PSZ      = B * CH * NC;         // 147,456
constexpr int EMAP_OFF = SRCP_OFF + PSZ;
constexpr int ACC_OFF  = EMAP_OFF + PSZ;      // 16 floats of accumulators
}

typedef __attribute__((ext_vector_type(2))) float v2f;
typedef __attribute__((ext_vector_type(8))) float v8f;

#if __has_builtin(__builtin_amdgcn_global_load_async_to_lds_b32) && \
    __has_builtin(__builtin_amdgcn_s_wait_asynccnt)
#define USE_ASYNC_LDS 1
typedef __attribute__((address_space(1))) int glob_i32;   // global (AS1) int
typedef __attribute__((address_space(3))) int lds_i32;    // LDS (AS3) int
#endif

// ---------------------------------------------------------------------------
// 0) zero accumulators
// ---------------------------------------------------------------------------
__global__ void zero_acc_kernel(float* __restrict__ acc) {
    if (threadIdx.x < 16) acc[threadIdx.x] = 0.0f;
}

// ---------------------------------------------------------------------------
// 1) softmax over 19 classes, one thread per pixel
// ---------------------------------------------------------------------------
__global__ __launch_bounds__(256)
void softmax_kernel(const float* __restrict__ logits, float* __restrict__ prob) {
    int idx = blockIdx.x * blockDim.x + threadIdx.x;
    if (idx >= NPIX) return;
    int b  = idx / (H * W);
    int hw = idx - b * (H * W);
    const float* L = logits + (size_t)b * C * H * W + hw;
    float*       P = prob   + (size_t)b * C * H * W + hw;
    float e[C];
    float mx = -3.0e38f;
#pragma unroll
    for (int c = 0; c < C; ++c) { e[c] = L[(size_t)c * H * W]; mx = fmaxf(mx, e[c]); }
    float s = 0.0f;
#pragma unroll
    for (int c = 0; c < C; ++c) { e[c] = expf(e[c] - mx); s += e[c]; }
    float inv = 1.0f / s;
#pragma unroll
    for (int c = 0; c < C; ++c) P[(size_t)c * H * W] = e[c] * inv;
}

// ---------------------------------------------------------------------------
// 2) 1x1 projection conv as GEMM via V_WMMA_F32_16X16X4_F32.
//    Y[b, m, n] = sum_k Wm[m,k] * X[b,k,n] + bias[m],  m<32, k<256, n<2304.
//    One wave per 16x16 output tile; 8 waves (tiles) per 256-thread block.
//    The A panel (16 rows x 256 K of proj_w, 16 KB) is staged into LDS once
//    per block with GLOBAL_LOAD_ASYNC_TO_LDS_B32 (ASYNCcnt) and then fed to
//    the WMMA from ds_load.
// ---------------------------------------------------------------------------
__global__ __launch_bounds__(256)
void proj_wmma_kernel(const float* __restrict__ X,   // [B, 256, 2304]
                      const float* __restrict__ Wm,  // [32, 256]
                      const float* __restrict__ bias,// [32]
                      float* __restrict__ Y)         // [B, 32, 2304]
{
    __shared__ float Alds[16 * KIN];                 // 16 KB

    const int lane = threadIdx.x & 31;
    const int wave = threadIdx.x >> 5;          // 0..7
    const int half = lane >> 4;                 // 0/1
    const int l    = lane & 15;
    const int n0 = (blockIdx.x * 8 + wave) * 16;   // 18 blocks * 8 waves * 16 = 2304
    const int m0 = blockIdx.y * 16;                // 0 or 16
    const int b  = blockIdx.z;
    const float* Xb = X + (size_t)b * KIN * NC;
    float*       Yb = Y + (size_t)b * CH * NC;

    // ---- stage A panel: rows m0..m0+15, all 256 K values -> LDS ----
#if defined(USE_ASYNC_LDS)
#pragma unroll
    for (int i = 0; i < 16; ++i) {
        const int e = threadIdx.x + i * 256;          // 0..4095 = row*256 + col
        __builtin_amdgcn_global_load_async_to_lds_b32(
            (glob_i32*)(Wm + m0 * KIN + e),
            (lds_i32*)&Alds[e], 0, 0);
    }
    __builtin_amdgcn_s_wait_asynccnt(0);
    __syncthreads();
#else
#pragma unroll
    for (int i = 0; i < 16; ++i) {
        const int e = threadIdx.x + i * 256;
        Alds[e] = Wm[m0 * KIN + e];
    }
    __syncthreads();
#endif

    v8f acc;
#pragma unroll
    for (int r = 0; r < 8; ++r) acc[r] = bias[m0 + r + 8 * half];   // fold bias into C

#if __has_builtin(__builtin_amdgcn_wmma_f32_16x16x4_f32)
#pragma unroll 4
    for (int k0 = 0; k0 < KIN; k0 += 4) {
        const int ka = k0 + 2 * half;
        v2f a, bb;
        // A 16x4 f32: lanes 0-15 K={0,1}, lanes 16-31 K={2,3}; row = lane&15
        a[0] = Alds[l * KIN + ka];
        a[1] = Alds[l * KIN + ka + 1];
        // B 4x16 f32: same K striping, N = lane&15
        bb[0] = Xb[(size_t)ka * NC + n0 + l];
        bb[1] = Xb[(size_t)(ka + 1) * NC + n0 + l];
        acc = __builtin_amdgcn_wmma_f32_16x16x4_f32(
                  false, a, false, bb, (short)0, acc, false, false);
    }
#else
    // scalar fallback with identical D layout
    for (int k = 0; k < KIN; ++k) {
        float xv = Xb[(size_t)k * NC + n0 + l];
#pragma unroll
        for (int r = 0; r < 8; ++r)
            acc[r] = fmaf(Alds[(r + 8 * half) * KIN + k], xv, acc[r]);
    }
#endif
    // D layout: VGPR r holds M = r + 8*half, N = lane&15
#pragma unroll
    for (int r = 0; r < 8; ++r)
        Yb[(size_t)(m0 + r + 8 * half) * NC + n0 + l] = acc[r];
}

// ---------------------------------------------------------------------------
// helper: dynamically select from a 9-entry register array with cndmask chain
// (keeps everything in VGPRs, no scratch)
// ---------------------------------------------------------------------------
__device__ __forceinline__ float sel9(const float v[9], int cell) {
    float r = v[0];
#pragma unroll
    for (int i = 1; i < 9; ++i) r = (cell == i) ? v[i] : r;
    return r;
}

// ---------------------------------------------------------------------------
// 3) fused loss kernel: one thread per fine pixel.
//    acc[0..2] = src_sim pos  {sum, cnt, sumsq}
//    acc[3..5] = src_sim neg  {sum, cnt, sumsq}
//    acc[6]    = sum loc_pos over masked pixels (9 terms each)
//    acc[7]    = sum loc_neg over masked pixels (8 terms each)
//    acc[8]    = masked pixel count
// ---------------------------------------------------------------------------
__global__ __launch_bounds__(256)
void loss_kernel(const float* __restrict__ prob,
                 const float* __restrict__ srcp,
                 const float* __restrict__ emap,
                 const int*   __restrict__ gt,
                 const int*   __restrict__ mix,
                 float* __restrict__ acc)
{
    int idx = blockIdx.x * blockDim.x + threadIdx.x;
    float part[9];
#pragma unroll
    for (int i = 0; i < 9; ++i) part[i] = 0.0f;

    if (idx < NPIX) {
        const int w = idx % W;
        const int h = (idx / W) % H;
        const int b = idx / (H * W);
        const int chh = h >> 2, cww = w >> 2;       // nearest-upsample factor 4

        // ---- 3x3 coarse squared feature distances -> similarities ----
        // (zero-padded outside: OOB neighbor feature = 0 -> d = ||f_center||^2)
        float se_s[9], se_e[9];
#pragma unroll
        for (int which = 0; which < 2; ++which) {
            const float* F = (which == 0 ? srcp : emap) + (size_t)b * CH * NC;
            float* se = (which == 0) ? se_s : se_e;
            float fc[CH];
#pragma unroll
            for (int c = 0; c < CH; ++c) fc[c] = F[(c * HC + chh) * WC + cww];
#pragma unroll
            for (int oi = -1; oi <= 1; ++oi) {
#pragma unroll
                for (int oj = -1; oj <= 1; ++oj) {
                    const int ni = chh + oi, nj = cww + oj;
                    float d = 0.0f;
                    if (ni >= 0 && ni < HC && nj >= 0 && nj < WC) {
                        const float* Fn = F + ni * WC + nj;
#pragma unroll
                        for (int c = 0; c < CH; ++c) {
                            float v = Fn[(size_t)c * NC] - fc[c];
                            d = fmaf(v, v, d);
                        }
                    } else {
#pragma unroll
                        for (int c = 0; c < CH; ++c) d = fmaf(fc[c], fc[c], d);
                    }
                    // only 9 distinct exps per tensor (instead of 25 per tap)
                    se[(oi + 1) * 3 + (oj + 1)] = expf(-d * INV_SIG2);
                }
            }
        }

        // ---- center class probs + gt ----
        const float* Pb = prob + (size_t)b * C * H * W;
        float pc[C];
#pragma unroll
        for (int c = 0; c < C; ++c) pc[c] = Pb[(size_t)c * H * W + h * W + w];
        const int gtc = gt[(b * H + h) * W + w];

        const int rr = h & 3, ss = w & 3;
        float sime[K2], crp[K2];
        bool  trg_ok = true;
        float sp_sum = 0, sp_cnt = 0, sp_sq = 0;
        float sn_sum = 0, sn_cnt = 0, sn_sq = 0;

#pragma unroll
        for (int t = 0; t < K2; ++t) {
            const int di = (t / KK - 2) * DILN;
            const int dj = (t % KK - 2) * DILN;
            // map fine tap -> coarse 3x3 cell (arithmetic shift: floor div)
            const int oi = ((rr + di) >> 2) + 1;
            const int oj = ((ss + dj) >> 2) + 1;
            const int cell = oi * 3 + oj;
            sime[t] = sel9(se_e, cell);
            const float ssim = sel9(se_s, cell);

            const int hh = h + di, ww = w + dj;
            const bool inb = (hh >= 0) & (hh < H) & (ww >= 0) & (ww < W);
            int   gtn = 0;
            float cp  = 0.0f;
            if (inb) {
                gtn = gt[(b * H + hh) * W + ww];
                const float* Pn = Pb + hh * W + ww;
#pragma unroll
                for (int c = 0; c < C; ++c) cp = fmaf(pc[c], Pn[(size_t)c * H * W], cp);
                if (mix[(b * H + hh) * W + ww] != 0) trg_ok = false;
            } else {
                trg_ok = false;     // zero-padded keep-tap -> sum < 25
            }
            crp[t] = cp;
            // ignore_src is all-true (labels < 255)
            if (gtn == gtc) { sp_sum += ssim; sp_cnt += 1.0f; sp_sq = fmaf(ssim, ssim, sp_sq); }
            else            { sn_sum += ssim; sn_cnt += 1.0f; sn_sq = fmaf(ssim, ssim, sn_sq); }
        }

        // ---- top-(K+1) largest / top-K smallest of ema_sim, jax tie order ----
        float simpos = 0.0f, simneg = 0.0f;
        {
            unsigned used = 0u;
#pragma unroll
            for (int k = 0; k < TOPK + 1; ++k) {
                float best = -3.0e38f, bc = 0.0f; int bi = 0;
#pragma unroll
                for (int t = 0; t < K2; ++t) {
                    const bool take = (((used >> t) & 1u) == 0u) && (sime[t] > best);
                    best = take ? sime[t] : best;
                    bc   = take ? crp[t]  : bc;
                    bi   = take ? t       : bi;
                }
                used |= (1u << bi);
                simpos = fmaf(best, -bc, simpos);               // max_v * -pos_g
            }
        }
        {
            unsigned used = 0u;
#pragma unroll
            for (int k = 0; k < TOPK; ++k) {
                float best = 3.0e38f, bc = 0.0f; int bi = 0;
#pragma unroll
                for (int t = 0; t < K2; ++t) {
                    const bool take = (((used >> t) & 1u) == 0u) && (sime[t] < best);
                    best = take ? sime[t] : best;
                    bc   = take ? crp[t]  : bc;
                    bi   = take ? t       : bi;
                }
                used |= (1u << bi);
                simneg = fmaf(1.0f - best, -(1.0f - bc), simneg); // (1-min_v) * -neg_g
            }
        }

        const float mk = trg_ok ? 1.0f : 0.0f;
        part[0] = sp_sum;      part[1] = sp_cnt;      part[2] = sp_sq;
        part[3] = sn_sum;      part[4] = sn_cnt;      part[5] = sn_sq;
        part[6] = simpos * mk; part[7] = simneg * mk; part[8] = mk;
    }

    // ---- wave32 shuffle reduction, one atomicAdd per wave per accumulator ----
    const int lane = threadIdx.x & 31;
#pragma unroll
    for (int i = 0; i < 9; ++i) {
        float v = part[i];
#pragma unroll
        for (int off = 16; off > 0; off >>= 1) v += __shfl_down(v, off, 32);
        if (lane == 0) atomicAdd(&acc[i], v);
    }
}

// ---------------------------------------------------------------------------
// 4) finalize: 9 accumulators -> 6 losses
// ---------------------------------------------------------------------------
__global__ void finalize_kernel(const float* __restrict__ acc, float* __restrict__ out) {
    if (threadIdx.x != 0 || blockIdx.x != 0) return;
    const float np = acc[1], nn = acc[4];
    const float mup = acc[0] / np;
    const float mun = acc[3] / nn;
    const float varp = (acc[2] - np * mup * mup) / (np - 1.0f);   // unbiased
    const float varn = (acc[5] - nn * mun * mun) / (nn - 1.0f);
    out[0] = -mup * 1.0f;                       // W_SRC_POS
    out[1] =  mun * 1.0f;                       // W_SRC_NEG
    out[2] = sqrtf(fmaxf(varp, 0.0f)) * 0.1f;   // W_SRC_POS_STD
    out[3] = sqrtf(fmaxf(varn, 0.0f)) * 0.1f;   // W_SRC_NEG_STD
    out[4] = acc[6] / (9.0f * acc[8]);          // mask broadcast over TOPK+1
    out[5] = acc[7] / (8.0f * acc[8]);          // mask broadcast over TOPK
}

// ---------------------------------------------------------------------------
extern "C" void kernel_launch(void* const* d_in, const int* in_sizes, int n_in,
                              void* d_out, int out_size, void* d_ws, size_t ws_size,
                              hipStream_t stream) {
    const float* logits_trg = (const float*)d_in[0];
    // d_in[1] (logits_ema) is unused by the reference
    const int*   gt_src     = (const int*)d_in[2];
    const int*   mix_masks  = (const int*)d_in[3];
    const float* x_ema      = (const float*)d_in[4];
    const float* x_src      = (const float*)d_in[5];
    const float* proj_w     = (const float*)d_in[6];
    const float* proj_b     = (const float*)d_in[7];

    float* ws   = (float*)d_ws;
    float* prob = ws + PROB_OFF;
    float* srcp = ws + SRCP_OFF;
    float* emap = ws + EMAP_OFF;
    float* acc  = ws + ACC_OFF;

    zero_acc_kernel<<<1, 32, 0, stream>>>(acc);
    softmax_kernel<<<(NPIX + 255) / 256, 256, 0, stream>>>(logits_trg, prob);
    proj_wmma_kernel<<<dim3(18, 2, B), 256, 0, stream>>>(x_src, proj_w, proj_b, srcp);
    proj_wmma_kernel<<<dim3(18, 2, B), 256, 0, stream>>>(x_ema, proj_w, proj_b, emap);
    loss_kernel<<<(NPIX + 255) / 256, 256, 0, stream>>>(prob, srcp, emap,
                                                        gt_src, mix_masks, acc);
    finalize_kernel<<<1, 32, 0, stream>>>(acc, (float*)d_out);
}